// DeepOKAN_38044820308187
// MI455X (gfx1250) — compile-verified
//
#include <hip/hip_runtime.h>
#include <hip/hip_bf16.h>

// DeepOKAN fused DeepONet-diag kernel set for gfx1250 (MI455X).
//
// Roofline: dominant cost is streaming W_b2 (2048 x 65536 fp32 = 512 MB)
// exactly once -> ~22 us at 23.3 TB/s. The diagonal fusion
//   V[j,n] = sum_k W_b2[j, n*64+k] * y_trunk[n,k]
//   Y[b,n] = sum_j basis2[b,j] * V[j,n]
// cuts FLOPs from 17.5 GF to ~0.8 GF, so the run is bandwidth-bound and we
// keep everything fp32. The final [64,2048]@[2048,1024] GEMM uses
// V_WMMA_F32_16X16X4_F32 (wave32 WMMA).

#define GRID_COUNT 8
#define SENSOR_DIM 1024
#define HD 64
#define BATCH 64
#define N_POINTS 1024
#define BRANCH_HIDDEN 256
#define TRUNK_HIDDEN 128

typedef __attribute__((ext_vector_type(2))) float v2f;
typedef __attribute__((ext_vector_type(8))) float v8f;

// RBF over grid [-1,1], 8 points -> spacing h = 2/7, inv = 3.5
__device__ __forceinline__ float rbf(float x, int g) {
    float c = -1.0f + (2.0f / 7.0f) * (float)g;
    float d = (x - c) * 3.5f;
    return __expf(-d * d);
}

// ---------------------------------------------------------------------------
// Trunk layer 1: x_trunk[1024,2] -> bt2[1024, 1024] (basis of hidden, fused)
// one block per point n, 128 threads (one per hidden unit o)
// ---------------------------------------------------------------------------
__global__ void trunk1_kernel(const float* __restrict__ xt,
                              const float* __restrict__ Wt1,
                              float* __restrict__ bt2) {
    int n = blockIdx.x;
    int o = threadIdx.x;                      // 0..127
    __shared__ float bas[16];                 // 2 coords x 8 grid
    if (o < 16) {
        int c = o >> 3, g = o & 7;
        bas[o] = rbf(xt[n * 2 + c], g);
    }
    __syncthreads();
    float acc = 0.0f;
#pragma unroll
    for (int r = 0; r < 16; ++r)
        acc = fmaf(bas[r], Wt1[r * TRUNK_HIDDEN + o], acc);
    // emit RBF basis of hidden activation directly (layer-2 input)
#pragma unroll
    for (int g = 0; g < GRID_COUNT; ++g)
        bt2[n * (TRUNK_HIDDEN * GRID_COUNT) + o * GRID_COUNT + g] = rbf(acc, g);
}

// ---------------------------------------------------------------------------
// Trunk layer 2: bt2[1024,1024] @ W_t2[1024,64] -> y_trunk[1024,64]
// one block per point n, 64 threads (one per HD channel k)
// ---------------------------------------------------------------------------
__global__ void trunk2_kernel(const float* __restrict__ bt2,
                              const float* __restrict__ Wt2,
                              float* __restrict__ ytr) {
    int n = blockIdx.x;
    int k = threadIdx.x;                      // 0..63
    __shared__ float row[TRUNK_HIDDEN * GRID_COUNT];  // 1024 floats
    for (int r = k; r < TRUNK_HIDDEN * GRID_COUNT; r += 64)
        row[r] = bt2[n * (TRUNK_HIDDEN * GRID_COUNT) + r];
    __syncthreads();
    float acc = 0.0f;
#pragma unroll 8
    for (int r = 0; r < TRUNK_HIDDEN * GRID_COUNT; ++r)
        acc = fmaf(row[r], Wt2[r * HD + k], acc);
    ytr[n * HD + k] = acc;
}

// ---------------------------------------------------------------------------
// Branch layer 1 (fused with layer-2 basis):
//   basis(x_branch[b,:])[8192] @ W_b1[8192,256] -> h1[256] -> basis2[b,2048]
// one block per batch row b, 256 threads (one per hidden unit o)
// ---------------------------------------------------------------------------
__global__ void branch1_kernel(const float* __restrict__ xb,
                               const float* __restrict__ Wb1,
                               float* __restrict__ basis2) {
    int b = blockIdx.x;
    int o = threadIdx.x;                      // 0..255
    __shared__ float bas[SENSOR_DIM * GRID_COUNT];    // 32 KB of the 320 KB WGP LDS
    for (int i = o; i < SENSOR_DIM; i += 256) {
        float x = xb[b * SENSOR_DIM + i];
#pragma unroll
        for (int g = 0; g < GRID_COUNT; ++g)
            bas[i * GRID_COUNT + g] = rbf(x, g);
    }
    __syncthreads();
    float acc = 0.0f;
#pragma unroll 8
    for (int r = 0; r < SENSOR_DIM * GRID_COUNT; ++r)   // LDS broadcast + coalesced Wb1
        acc = fmaf(bas[r], Wb1[(size_t)r * BRANCH_HIDDEN + o], acc);
    // emit layer-2 RBF basis directly: basis2[b, o*8+g]
#pragma unroll
    for (int g = 0; g < GRID_COUNT; ++g)
        basis2[b * (BRANCH_HIDDEN * GRID_COUNT) + o * GRID_COUNT + g] = rbf(acc, g);
}

// ---------------------------------------------------------------------------
// V[j,n] = sum_{k<64} W_b2[j, n*64+k] * y_trunk[n,k]
// The bandwidth-dominant kernel: streams all 512 MB of W_b2 once.
// one block per point n (1024 blocks), 8 waves; wave w handles rows j = w mod 8.
// Each wave reads a contiguous 256 B chunk (float2/lane) -> full-line coalescing.
// ---------------------------------------------------------------------------
__global__ void v_kernel(const float* __restrict__ Wb2,
                         const float* __restrict__ ytr,
                         float* __restrict__ V) {
    int n = blockIdx.x;
    int wave = threadIdx.x >> 5;              // 0..7
    int lane = threadIdx.x & 31;
    __shared__ float tn[HD];
    if (threadIdx.x < HD) tn[threadIdx.x] = ytr[n * HD + threadIdx.x];
    __syncthreads();
    float t0 = tn[2 * lane];
    float t1 = tn[2 * lane + 1];
    const size_t rowStride = (size_t)SENSOR_DIM * HD;   // 65536 floats
    for (int j = wave; j < BRANCH_HIDDEN * GRID_COUNT; j += 8) {
        const float* p = Wb2 + (size_t)j * rowStride + (size_t)n * HD + 2 * lane;
        if (j + 8 < BRANCH_HIDDEN * GRID_COUNT)
            __builtin_prefetch(p + 8 * rowStride, 0, 0);   // global_prefetch_b8
        float2 w = *reinterpret_cast<const float2*>(p);
        float s = fmaf(w.x, t0, w.y * t1);
        // wave32 butterfly reduction
#pragma unroll
        for (int m = 16; m >= 1; m >>= 1)
            s += __shfl_xor(s, m, 32);
        if (lane == 0) V[(size_t)j * N_POINTS + n] = s;
    }
}

// ---------------------------------------------------------------------------
// Y[b,n] = sum_{j<2048} basis2[b,j] * V[j,n]   ([64,2048] @ [2048,1024])
// V_WMMA_F32_16X16X4_F32, one wave per 16x16 output tile, K-loop of 512 WMMAs.
// A(16x4 fp32): v2f/lane; lanes 0-15 carry K={k,k+1}, lanes 16-31 K={k+2,k+3}.
// C/D(16x16 fp32): v8f; VGPR r holds M=r (lanes 0-15) / M=r+8 (lanes 16-31).
// ---------------------------------------------------------------------------
__global__ void y_wmma_kernel(const float* __restrict__ basis2,
                              const float* __restrict__ V,
                              float* __restrict__ out) {
    const int K = BRANCH_HIDDEN * GRID_COUNT;   // 2048
    int tile = blockIdx.x;                      // 256 tiles = 4 (M) x 64 (N)
    int mt = tile & 3;
    int nt = tile >> 2;
    int l = threadIdx.x;                        // 0..31, wave32, EXEC all ones
    int hi = (l >= 16) ? 1 : 0;
    int ml = l & 15;
    int rowA = mt * 16 + ml;                    // batch row
    int colB = nt * 16 + ml;                    // point column
    const float* aRow = basis2 + (size_t)rowA * K;
    v8f c = {};
    for (int k = 0; k < K; k += 4) {
        int ka = k + (hi ? 2 : 0);
        v2f a, b;
        a.x = aRow[ka];
        a.y = aRow[ka + 1];
        b.x = V[(size_t)ka * N_POINTS + colB];
        b.y = V[(size_t)(ka + 1) * N_POINTS + colB];
        // (neg_a, A, neg_b, B, c_mod, C, reuse_a, reuse_b)
        c = __builtin_amdgcn_wmma_f32_16x16x4_f32(false, a, false, b,
                                                  (short)0, c, false, false);
    }
#pragma unroll
    for (int r = 0; r < 8; ++r) {
        int m = r + (hi ? 8 : 0);
        out[(size_t)(mt * 16 + m) * N_POINTS + nt * 16 + ml] = c[r];
    }
}

// ---------------------------------------------------------------------------
extern "C" void kernel_launch(void* const* d_in, const int* in_sizes, int n_in,
                              void* d_out, int out_size, void* d_ws, size_t ws_size,
                              hipStream_t stream) {
    const float* x_branch = (const float*)d_in[0];   // [64,1024]
    const float* x_trunk  = (const float*)d_in[1];   // [1024,2]
    const float* W_b1     = (const float*)d_in[2];   // [1024,8,256]
    const float* W_b2     = (const float*)d_in[3];   // [256,8,65536]
    const float* W_t1     = (const float*)d_in[4];   // [2,8,128]
    const float* W_t2     = (const float*)d_in[5];   // [128,8,64]
    float* out = (float*)d_out;                      // [64,1024,1]

    // workspace carve-out (~12.75 MB total)
    float* bt2    = (float*)d_ws;                          // 1024*1024  (4 MB)
    float* ytr    = bt2    + (size_t)N_POINTS * TRUNK_HIDDEN * GRID_COUNT; // 1024*64
    float* basis2 = ytr    + (size_t)N_POINTS * HD;        // 64*2048
    float* V      = basis2 + (size_t)BATCH * BRANCH_HIDDEN * GRID_COUNT;   // 2048*1024 (8 MB)

    trunk1_kernel <<<N_POINTS, TRUNK_HIDDEN, 0, stream>>>(x_trunk, W_t1, bt2);
    trunk2_kernel <<<N_POINTS, HD,           0, stream>>>(bt2, W_t2, ytr);
    branch1_kernel<<<BATCH,    BRANCH_HIDDEN,0, stream>>>(x_branch, W_b1, basis2);
    v_kernel      <<<N_POINTS, 256,          0, stream>>>(W_b2, ytr, V);
    y_wmma_kernel <<<256,      32,           0, stream>>>(basis2, V, out);
}